// TriMipEncodingCUDA_81192061764393
// MI455X (gfx1250) — compile-verified
//
#include <hip/hip_runtime.h>

#define NLV 8
#define FD 16

typedef __attribute__((ext_vector_type(2))) float v2f;
typedef __attribute__((ext_vector_type(4))) float f4;
typedef __attribute__((ext_vector_type(8))) float v8f;

// float-offset of mip level l (l>=1) inside the mips workspace:
// 48 * sum_{j=1..l-1} (512>>j)^2  ==  16*(262144 - (1<<(20-2l)))
__host__ __device__ constexpr unsigned mip_off(int l) {
  return 16u * (262144u - (1u << (20 - 2 * l)));
}

// ---------------------------------------------------------------------------
// Mip 2x2-mean downsample as a sparse matmul on the WMMA pipe.
// One wave32 per tile: 16 output texels (one row segment) x 16 features.
// D(16x16) = sum_{k4=0..15} A_chunk(16x4) x B_chunk(4x16), A entries = 0.25.
// Inputs per tile: 2 rows x 32 cols of the source mip.
// ---------------------------------------------------------------------------
__global__ void downsample_wmma(const float* __restrict__ src,
                                float* __restrict__ dst, int S_out) {
  int gid  = blockIdx.x * blockDim.x + threadIdx.x;
  int wave = gid >> 5;
  int lane = threadIdx.x & 31;
  int tilesPerRow   = S_out >> 4;
  int tilesPerPlane = S_out * tilesPerRow;
  if (wave >= 3 * tilesPerPlane) return;          // wave-uniform exit, EXEC stays all-1s
  int plane = wave / tilesPerPlane;
  int rem   = wave - plane * tilesPerPlane;
  int y     = rem / tilesPerRow;
  int X0    = (rem - y * tilesPerRow) << 4;
  int S_in  = S_out << 1;
  const float* sp = src + (size_t)plane * S_in * S_in * FD;

  int n    = lane & 15;   // A: M-row / B: feature column / D: feature column
  int half = lane >> 4;   // K lane-half selector

  v8f c = {0.f, 0.f, 0.f, 0.f, 0.f, 0.f, 0.f, 0.f};
#pragma unroll
  for (int k4 = 0; k4 < 16; ++k4) {
    // global K = 4*k4 + (2*half + vgpr); input row r, col cb(+vgpr)
    int r  = k4 >> 3;
    int cb = ((k4 & 7) << 2) + (half << 1);
    const float* bp = sp + ((size_t)(2 * y + r) * S_in + (2 * X0 + cb)) * FD + n;
    v2f b;
    b.x = bp[0];      // B[k_local = 2*half+0][n]
    b.y = bp[FD];     // B[k_local = 2*half+1][n]
    // A[m][j] = 0.25 iff output texel m owns input col (k&31): m == 2*(k4&7)+half
    int mq = ((k4 & 7) << 1) + half;
    float av = (n == mq) ? 0.25f : 0.0f;
    v2f a;
    a.x = av;
    a.y = av;
    c = __builtin_amdgcn_wmma_f32_16x16x4_f32(false, a, false, b, (short)0, c,
                                              false, false);
  }

  // D layout: VGPR v, lanes 0-15 -> M=v ; lanes 16-31 -> M=8+v ; N = lane&15
  float* dp = dst + ((size_t)plane * S_out + y) * (size_t)S_out * FD;
#pragma unroll
  for (int v = 0; v < 8; ++v) {
    int m = half * 8 + v;
    dp[(size_t)(X0 + m) * FD + n] = c[v];
  }
}

// Tiny levels (8^2, 4^2): plain scalar downsample.
__global__ void downsample_scalar(const float* __restrict__ src,
                                  float* __restrict__ dst, int S_out) {
  int idx = blockIdx.x * blockDim.x + threadIdx.x;
  int total = 3 * S_out * S_out * FD;
  if (idx >= total) return;
  int f = idx & (FD - 1);
  int t = idx >> 4;
  int x = t % S_out; t /= S_out;
  int y = t % S_out;
  int plane = t / S_out;
  int S_in = S_out << 1;
  const float* sp =
      src + (((size_t)plane * S_in + 2 * y) * S_in + 2 * x) * FD + f;
  size_t rs = (size_t)S_in * FD;
  dst[idx] = 0.25f * (sp[0] + sp[FD] + sp[rs] + sp[rs + FD]);
}

// ---------------------------------------------------------------------------
// Sampling: one thread per (point, plane); 2 mip levels x 4 bilinear taps,
// each tap = one 64B-aligned cacheline (16 f32 features) = 4x b128 loads.
// ---------------------------------------------------------------------------
__device__ __forceinline__ void sample_add(const float* __restrict__ tex, int S,
                                           float u, float v, float w,
                                           f4 acc[4]) {
  float px = u * (float)S - 0.5f;
  float py = v * (float)S - 0.5f;
  float fx0 = floorf(px), fy0 = floorf(py);
  float fx = px - fx0, fy = py - fy0;
  int ix0 = (int)fx0, iy0 = (int)fy0;
  int x0 = min(max(ix0, 0), S - 1);
  int y0 = min(max(iy0, 0), S - 1);
  int x1 = min(max(ix0 + 1, 0), S - 1);
  int y1 = min(max(iy0 + 1, 0), S - 1);
  float w00 = w * (1.f - fx) * (1.f - fy);
  float w10 = w * fx * (1.f - fy);
  float w01 = w * (1.f - fx) * fy;
  float w11 = w * fx * fy;
  const f4* t00 = (const f4*)(tex + ((size_t)y0 * S + x0) * FD);
  const f4* t10 = (const f4*)(tex + ((size_t)y0 * S + x1) * FD);
  const f4* t01 = (const f4*)(tex + ((size_t)y1 * S + x0) * FD);
  const f4* t11 = (const f4*)(tex + ((size_t)y1 * S + x1) * FD);
#pragma unroll
  for (int q = 0; q < 4; ++q) {
    acc[q] += w00 * t00[q] + w10 * t10[q] + w01 * t01[q] + w11 * t11[q];
  }
}

__global__ void trimip_sample(const float* __restrict__ xs,
                              const float* __restrict__ level,
                              const float* __restrict__ fm,
                              const float* __restrict__ mips,
                              float* __restrict__ out, int npts) {
  int tid = blockIdx.x * blockDim.x + threadIdx.x;
  if (tid >= npts * 3) return;
  int point = tid / 3;
  int plane = tid - point * 3;
  float xc0 = xs[point * 3 + 0];
  float xc1 = xs[point * 3 + 1];
  float xc2 = xs[point * 3 + 2];
  float u = (plane == 0) ? xc1 : xc0;
  float v = (plane == 2) ? xc1 : xc2;

  float lvl = fminf(fmaxf(level[point], 0.0f), 7.0f);
  float l0f = floorf(lvl);
  int l0 = (int)l0f;
  float fr = lvl - l0f;         // weight of level l0+1; (1-fr) for l0
  int l1 = min(l0 + 1, 7);      // when l0==7, fr==0 so second term vanishes

  f4 acc[4] = {};

  int S0 = 512 >> l0;
  const float* t0 =
      (l0 == 0) ? fm + (size_t)plane * (512 * 512 * FD)
                : mips + mip_off(l0) + (size_t)plane * S0 * S0 * FD;
  sample_add(t0, S0, u, v, 1.0f - fr, acc);

  int S1 = 512 >> l1;  // l1 >= 1 always
  const float* t1 = mips + mip_off(l1) + (size_t)plane * S1 * S1 * FD;
  sample_add(t1, S1, u, v, fr, acc);

  // 384MB write-once stream: non-temporal so the mips stay resident in L2
  f4* o = (f4*)(out + (size_t)tid * FD);
#pragma unroll
  for (int q = 0; q < 4; ++q) __builtin_nontemporal_store(acc[q], o + q);
}

// ---------------------------------------------------------------------------
extern "C" void kernel_launch(void* const* d_in, const int* in_sizes, int n_in,
                              void* d_out, int out_size, void* d_ws,
                              size_t ws_size, hipStream_t stream) {
  const float* xs    = (const float*)d_in[0];
  const float* level = (const float*)d_in[1];
  const float* fm    = (const float*)d_in[2];
  float* mips = (float*)d_ws;     // needs 4,194,048 floats (~16.8 MB)
  float* out  = (float*)d_out;
  int npts = in_sizes[0] / 3;

  // Build mip chain 1..7 (sequential 2x2 means, matching _build_mips)
  const float* src = fm;
  for (int l = 1; l < NLV; ++l) {
    int S_out = 512 >> l;
    float* dst = mips + mip_off(l);
    if (S_out >= 16) {
      int nTiles = 3 * S_out * (S_out >> 4);
      int nBlocks = (nTiles + 7) / 8;  // 8 wave32 per 256-thread block
      downsample_wmma<<<nBlocks, 256, 0, stream>>>(src, dst, S_out);
    } else {
      int total = 3 * S_out * S_out * FD;
      downsample_scalar<<<(total + 255) / 256, 256, 0, stream>>>(src, dst,
                                                                 S_out);
    }
    src = dst;
  }

  int total = npts * 3;
  trimip_sample<<<(total + 255) / 256, 256, 0, stream>>>(xs, level, fm, mips,
                                                         out, npts);
}